// ViolationInformedLoss_63591285784812
// MI455X (gfx1250) — compile-verified
//
#include <hip/hip_runtime.h>
#include <math.h>

typedef __attribute__((ext_vector_type(2))) float v2f;
typedef __attribute__((ext_vector_type(8))) float v8f;

#define BATCH   2048
#define KDIM    256
#define PDIM    256
#define SNUM    10
#define NMAX    512      // KDIM + PDIM
#define SROWS   16       // sample rows padded to WMMA M=16
#define EPSV    1e-9f
#define LAMBDAV 0.5f

// One workgroup per batch element. 8 waves; each wave owns 32 P-columns
// (two 16x16 WMMA N-tiles), accumulating C2 = Xm @ Pm over K=256 with
// V_WMMA_F32_16X16X4_F32 (full f32 precision, matches the f32 reference).
__global__ __launch_bounds__(256)
void vil_batch_kernel(const float* __restrict__ y_pred,
                      const float* __restrict__ y_true,
                      const float* __restrict__ Pmat,
                      const float* __restrict__ params,
                      const float* __restrict__ Xg,
                      float* __restrict__ ws)
{
    __shared__ float Xs[SROWS * KDIM];     // 16 KB: masked, zero-padded A tile
    __shared__ float mags[SNUM * NMAX];    // 20 KB: |C| with -1 sentinels
    __shared__ float hmbuf[16];

    const int b   = blockIdx.x;
    const int tid = threadIdx.x;

    const int n_par = (int)params[b * 3 + 0];
    const int k_par = (int)params[b * 3 + 1];
    const int m_par = (int)params[b * 3 + 2];
    const int nk    = n_par - k_par;       // valid P-columns (pmask)

    const float* __restrict__ Xb = Xg   + (size_t)b * (SNUM * KDIM);
    const float* __restrict__ Pb = Pmat + (size_t)b * (KDIM * PDIM);

    // ---- stage Xm into LDS: rows >= SNUM and cols >= k zeroed (xmask) ----
    for (int idx = tid; idx < SROWS * KDIM; idx += 256) {
        const int s  = idx >> 8;           // row
        const int kk = idx & 255;          // col
        float v = 0.0f;
        if (s < SNUM && kk < k_par) v = Xb[s * KDIM + kk];
        Xs[idx] = v;
    }
    __syncthreads();

    // ---- C2 = A(16x256) * P(256x256), f32 WMMA, K step = 4 ----
    const int wave = tid >> 5;
    const int lane = tid & 31;
    const int lhi  = lane >> 4;            // 0 | 1  (K half)
    const int llo  = lane & 15;            // M for A-frag, N for B-frag
    const int n0   = wave * 32;            // this wave's P-column base

    v8f acc0 = {0.f, 0.f, 0.f, 0.f, 0.f, 0.f, 0.f, 0.f};
    v8f acc1 = acc0;

    for (int k0 = 0; k0 < KDIM; k0 += 4) {
        const int krow = k0 + (lhi << 1);
        // A 16x4 f32 layout: lane llo = M; VGPR0/1 = K = krow, krow+1
        v2f a = *(const v2f*)(&Xs[llo * KDIM + krow]);   // ds_load_b64
        // B 4x16 f32 layout (mirror of A): lane llo = N
        v2f b0, b1;
        b0.x = Pb[(krow    ) * PDIM + n0 + llo];
        b0.y = Pb[(krow + 1) * PDIM + n0 + llo];
        b1.x = Pb[(krow    ) * PDIM + n0 + 16 + llo];
        b1.y = Pb[(krow + 1) * PDIM + n0 + 16 + llo];
        acc0 = __builtin_amdgcn_wmma_f32_16x16x4_f32(
            false, a, false, b0, (short)0, acc0, false, false);
        acc1 = __builtin_amdgcn_wmma_f32_16x16x4_f32(
            false, a, false, b1, (short)0, acc1, false, false);
    }

    // ---- |C2| with pmask -> mags cols [KDIM, KDIM+PDIM) ----
    // C/D f32 16x16 layout: VGPR r, lanes<16: M=r; lanes>=16: M=r+8; N=llo
    #pragma unroll
    for (int r = 0; r < 8; ++r) {
        const int mrow = r + (lhi << 3);
        if (mrow < SNUM) {
            const int c0 = n0 + llo;
            const int c1 = n0 + 16 + llo;
            mags[mrow * NMAX + KDIM + c0] = (c0 < nk) ? fabsf(acc0[r]) : -1.0f;
            mags[mrow * NMAX + KDIM + c1] = (c1 < nk) ? fabsf(acc1[r]) : -1.0f;
        }
    }

    // ---- |C1| (identity block) -> mags cols [0, KDIM) ----
    for (int idx = tid; idx < SNUM * KDIM; idx += 256) {
        const int s = idx >> 8;
        const int j = idx & 255;
        mags[s * NMAX + j] = (j < k_par) ? fabsf(Xs[s * KDIM + j]) : -1.0f;
    }
    __syncthreads();

    // ---- top-9 per sample row (m_par <= 8), insertion scan ----
    if (tid < SNUM) {
        float top[9];
        #pragma unroll
        for (int i = 0; i < 9; ++i) top[i] = -2.0f;
        const float* row = &mags[tid * NMAX];
        for (int j = 0; j < NMAX; ++j) {
            const float v = row[j];
            if (v > top[8]) {
                int p = 8;
                while (p > 0 && top[p - 1] < v) { top[p] = top[p - 1]; --p; }
                top[p] = v;
            }
        }
        hmbuf[tid] = top[0] / (top[m_par] + EPSV);
    }
    __syncthreads();

    // ---- per-batch penalty + log-mse term -> workspace ----
    if (tid == 0) {
        float mh = hmbuf[0];
        for (int s = 1; s < SNUM; ++s) mh = fmaxf(mh, hmbuf[s]);
        float pen = 0.0f;
        if (m_par + 1 <= n_par) pen = fmaxf(mh - y_pred[b], 0.0f);
        ws[b] = pen;
        const float lp = log2f(fmaxf(y_pred[b], EPSV));
        const float lt = log2f(fmaxf(y_true[b], EPSV));
        const float d  = lt - lp;
        ws[BATCH + b] = d * d;
    }
}

// Deterministic fixed-order reduction (no float atomics) -> 3 output scalars.
__global__ __launch_bounds__(256)
void vil_reduce_kernel(const float* __restrict__ ws, float* __restrict__ out)
{
    __shared__ float sp[256];
    __shared__ float sq[256];
    const int t = threadIdx.x;
    float ap = 0.0f, aq = 0.0f;
    for (int i = t; i < BATCH; i += 256) {
        ap += ws[i];
        aq += ws[BATCH + i];
    }
    sp[t] = ap; sq[t] = aq;
    __syncthreads();
    for (int off = 128; off > 0; off >>= 1) {
        if (t < off) { sp[t] += sp[t + off]; sq[t] += sq[t + off]; }
        __syncthreads();
    }
    if (t == 0) {
        const float viol   = sp[0] / (float)BATCH;
        const float logmse = sq[0] / (float)BATCH;
        out[0] = logmse + LAMBDAV * viol;   // total_loss
        out[1] = logmse;                    // loss_logmse
        out[2] = viol;                      // loss_violation
    }
}

extern "C" void kernel_launch(void* const* d_in, const int* in_sizes, int n_in,
                              void* d_out, int out_size, void* d_ws, size_t ws_size,
                              hipStream_t stream) {
    const float* y_pred = (const float*)d_in[0];
    const float* y_true = (const float*)d_in[1];
    const float* Pmat   = (const float*)d_in[2];
    const float* params = (const float*)d_in[3];
    const float* Xg     = (const float*)d_in[4];
    float* out = (float*)d_out;
    float* ws  = (float*)d_ws;     // needs 2*BATCH floats = 16 KB

    vil_batch_kernel<<<BATCH, 256, 0, stream>>>(y_pred, y_true, Pmat, params, Xg, ws);
    vil_reduce_kernel<<<1, 256, 0, stream>>>(ws, out);
}